// CurvatureField_35424890257807
// MI455X (gfx1250) — compile-verified
//
#include <hip/hip_runtime.h>

typedef __attribute__((ext_vector_type(2))) float v2f;
typedef __attribute__((ext_vector_type(8))) float v8f;

// One wave (32 threads) handles one batch b.
// LDS: Gamma[b] staged with row stride 17 (bank-conflict free for both the
// A-view gather (stride-16 rows) and B-view gather), augmented 16x33 matrix
// for Gauss-Jordan inverse, plus small reduction scratch. ~20.8 KB total.
__global__ __launch_bounds__(32) void CurvatureField_kernel(
    const float* __restrict__ metric,   // [B,16,16]
    const float* __restrict__ gamma,    // [B,16,16,16]
    float* __restrict__ out,            // [B]
    int Bn)
{
    __shared__ float sG[256 * 17];   // Gamma[b]: element (m,i,k) at (m*16+i)*17 + k
    __shared__ float sS1[256];       // S1[i*16+j]
    __shared__ float sS2[16];        // s2[i]
    __shared__ float sTT[16];        // t[m]
    __shared__ float sAug[16 * 33];  // [A | I], row stride 33
    __shared__ float sRed[32];

    const int b    = blockIdx.x;
    const int L    = threadIdx.x;   // 0..31
    const int half = L >> 4;        // 0 or 1
    const int j    = L & 15;        // col index (also A-row index i for A gather)
    const int kk0  = half * 2;      // K-slot pair: lanes 0-15 -> K=0,1 ; 16-31 -> K=2,3

    const float* G  = gamma + (size_t)b * 4096;
    const float* Gp = gamma + (size_t)((b == 0) ? (Bn - 1) : (b - 1)) * 4096;

    // ---- Phase 1: stage Gamma[b] to LDS; accumulate S1 = sum_k (Gp - G) ----
    // Linear element t = k*256 + (i*16+j); lane L exclusively owns plane
    // positions p = t & 255 with p ≡ L (mod 32)  (256 % 32 == 0).
    float acc[8];
#pragma unroll
    for (int w = 0; w < 8; ++w) acc[w] = 0.f;
    for (int u = 0; u < 128; ++u) {
        int   t  = u * 32 + L;
        float g  = G[t];
        float gp = Gp[t];
        sG[(t >> 4) * 17 + (t & 15)] = g;
        acc[u & 7] += gp - g;
    }
#pragma unroll
    for (int w = 0; w < 8; ++w) sS1[w * 32 + L] = acc[w];

    // init augmented [metric | I]; lane L owns column L for all 16 rows
    for (int r = 0; r < 16; ++r) {
        float v;
        if (L < 16) v = metric[(size_t)b * 256 + r * 16 + L];
        else        v = (r == (L - 16)) ? 1.0f : 0.0f;
        sAug[r * 33 + L] = v;
    }
    __syncthreads();

    // ---- Phase 2: s2[i] (lanes 0-15) and t[m] (lanes 16-31) ----
    if (half == 0) {
        const int i = j;
        float s = 0.f;
#pragma unroll
        for (int k = 0; k < 16; ++k) {
            int t = k * 256 + i * 16 + k;
            s += Gp[t] - sG[(t >> 4) * 17 + (t & 15)];
        }
        sS2[i] = s;
    } else {
        const int m = j;
        float s = 0.f;
#pragma unroll
        for (int k = 0; k < 16; ++k)
            s += sG[(k * 16 + m) * 17 + k];
        sTT[m] = s;
    }
    __syncthreads();

    // ---- Phase 3: C[i,j] = sum_{m,k} G[m,i,k]*G[k,m,j] via 64 x WMMA f32 16x16x4
    // Fused contraction index mk = m*16+k, chunked by 4. For chunk c:
    //   m = c>>2, k0 = (c&3)*4 + kk0 (and k0+1), so
    //   A[i,K]   = G[m, i, k]   (A layout: lanes0-15 K=0/1, lanes16-31 K=2/3)
    //   B[K,j]   = G[k, m, j]   (B layout mirrors A's K slotting)
    v8f C = {};
    for (int c = 0; c < 64; ++c) {
        int m  = c >> 2;
        int k0 = (c & 3) * 4 + kk0;
        v2f a, bb;
        a.x  = sG[(m * 16 + j) * 17 + k0];
        a.y  = sG[(m * 16 + j) * 17 + k0 + 1];
        bb.x = sG[(k0 * 16 + m) * 17 + j];
        bb.y = sG[((k0 + 1) * 16 + m) * 17 + j];
        C = __builtin_amdgcn_wmma_f32_16x16x4_f32(
                /*neg_a=*/false, a, /*neg_b=*/false, bb,
                /*c_mod=*/(short)0, C, /*reuse_a=*/false, /*reuse_b=*/false);
    }

    // ---- Phase 4: Gauss-Jordan inverse of SPD metric (no pivoting needed) ----
    for (int p = 0; p < 16; ++p) {
        float invP = 1.0f / sAug[p * 33 + p];
        sAug[p * 33 + L] *= invP;          // scale pivot row (lane L: column L)
        __syncthreads();
        float prow = sAug[p * 33 + L];
#pragma unroll
        for (int r = 0; r < 16; ++r) {
            if (r == p) continue;
            float f = sAug[r * 33 + p];    // wave-uniform broadcast read
            sAug[r * 33 + L] -= f * prow;  // lane-p write lands after all reads
        }
        __syncthreads();
    }
    // Minv[r][c] now at sAug[r*33 + 16 + c]

    // ---- Phase 5: ricci = S1 - s2 + r1 - C ; scalar = <ricci, Minv> ----
    // C/D layout: component v <-> (i = v + 8*half, j = L&15)
    float partial = 0.f;
#pragma unroll
    for (int v = 0; v < 8; ++v) {
        const int i = v + 8 * half;
        float r1 = 0.f;
#pragma unroll
        for (int m = 0; m < 16; ++m)
            r1 = fmaf(sTT[m], sG[(m * 16 + i) * 17 + j], r1);
        float ricci = sS1[i * 16 + j] - sS2[i] + r1 - C[v];
        partial += ricci * sAug[i * 33 + 16 + j];
    }
    sRed[L] = partial;
    __syncthreads();
    if (L == 0) {
        float s = 0.f;
#pragma unroll
        for (int q = 0; q < 32; ++q) s += sRed[q];
        out[b] = s;
    }
}

extern "C" void kernel_launch(void* const* d_in, const int* in_sizes, int n_in,
                              void* d_out, int out_size, void* d_ws, size_t ws_size,
                              hipStream_t stream) {
    const float* metric = (const float*)d_in[0];   // [B,16,16]
    const float* gamma  = (const float*)d_in[1];   // [B,16,16,16]
    float*       out    = (float*)d_out;           // [B]
    const int    Bn     = out_size;                // 2048

    CurvatureField_kernel<<<dim3(Bn), dim3(32), 0, stream>>>(metric, gamma, out, Bn);
}